// CPUTopKRouter_89799176225511
// MI455X (gfx1250) — compile-verified
//
#include <hip/hip_runtime.h>
#include <hip/hip_bf16.h>

typedef __attribute__((ext_vector_type(2))) float v2f;
typedef __attribute__((ext_vector_type(8))) float v8f;

#define Tdim 16384
#define Hdim 4096
#define Edim 64
#define TOPKK 8
#define KC   64     // K-chunk staged in LDS
#define LDA  68     // padded LDS stride (floats): 272B, 16B aligned, conflict-free
#define ROWS 32     // rows per block
#define LDE  66     // padded stride for logits stash

// CDNA5 async global->LDS copy (16B per lane), tracked by ASYNCcnt.
// %0 = LDS byte address (VGPR), %1 = 64-bit global address (VGPR pair).
__device__ __forceinline__ void async_copy_b128(unsigned lds, const float* g) {
    asm volatile("global_load_async_to_lds_b128 %0, %1, off"
                 :: "v"(lds), "v"(g) : "memory");
}
__device__ __forceinline__ void wait_async0() {
    asm volatile("s_wait_asynccnt 0x0" ::: "memory");
}

__global__ __launch_bounds__(256) void router_fused(
    const float* __restrict__ X, const float* __restrict__ W,
    float* __restrict__ logits, float* __restrict__ wout, int* __restrict__ iout)
{
    __shared__ float smem[2 * ROWS * LDA];   // 17408 B; reused for logits stash

    const int tid  = threadIdx.x;
    const int lane = tid & 31;
    const int wave = tid >> 5;
    const int row0 = blockIdx.x * ROWS;

    const int wrow = (wave >> 2) << 4;   // 0 or 16
    const int wcol = (wave & 3) << 4;    // 0,16,32,48

    const int lhalf = lane >> 4;         // 0: K+0.. / 1: K+2..
    const int lm    = lane & 15;

    // ---- chunk loader: 512 float4 per chunk, 2 async b128 per thread ----
    const int f0 = tid,        r0 = f0 >> 4, kv0 = (f0 & 15) << 2;
    const int f1 = tid + 256,  r1 = f1 >> 4, kv1 = (f1 & 15) << 2;
    const float* __restrict__ Xr0 = X + (size_t)(row0 + r0) * Hdim + kv0;
    const float* __restrict__ Xr1 = X + (size_t)(row0 + r1) * Hdim + kv1;

    // LDS byte addresses for this thread's two 16B destinations (buffer 0)
    const unsigned lds0 = (unsigned)(size_t)&smem[r0 * LDA + kv0];
    const unsigned lds1 = (unsigned)(size_t)&smem[r1 * LDA + kv1];
    const unsigned bufBytes = ROWS * LDA * 4;

    // B fragment pointer: lane n -> expert (wcol+n), consecutive k pairs
    const float* __restrict__ Wp = W + (size_t)(wcol + lm) * Hdim + (lhalf << 1);

    // A fragment base offset inside a buffer
    const int aoff = (wrow + lm) * LDA + (lhalf << 1);

    // prime buffer 0 via async DMA
    async_copy_b128(lds0, Xr0);
    async_copy_b128(lds1, Xr1);
    wait_async0();
    __syncthreads();

    v8f acc = {};
    unsigned buf = 0;
    for (int k0 = 0; k0 < Hdim; k0 += KC) {
        const bool more = (k0 + KC) < Hdim;
        if (more) {   // kick off next chunk into the other buffer (no VGPR staging)
            const unsigned nb = buf ^ 1;
            async_copy_b128(lds0 + nb * bufBytes, Xr0 + k0 + KC);
            async_copy_b128(lds1 + nb * bufBytes, Xr1 + k0 + KC);
        }
        const float* __restrict__ Ab = &smem[buf * (ROWS * LDA) + aoff];
        const float* __restrict__ Bb = Wp + k0;
#pragma unroll
        for (int kk = 0; kk < KC; kk += 4) {
            v2f a = *(const v2f*)(Ab + kk);   // ds_load_2addr_b64
            v2f b = *(const v2f*)(Bb + kk);   // global_load_b64 (L2-resident W)
            acc = __builtin_amdgcn_wmma_f32_16x16x4_f32(
                false, a, false, b, (short)0, acc, false, false);
        }
        if (more) {
            wait_async0();     // our copies landed in LDS
            __syncthreads();   // everyone's copies landed; reads of old buf done
            buf ^= 1;
        }
    }

    // ---- emit logits + stash tile in LDS for softmax ----
    __syncthreads();   // A buffers dead; reuse smem
    const int ccol = wcol + lm;
#pragma unroll
    for (int j = 0; j < 8; ++j) {
        const int r = wrow + (lhalf << 3) + j;   // C/D layout: vgpr j = row j / j+8
        const float v = acc[j];
        logits[(size_t)(row0 + r) * Edim + ccol] = v;
        smem[r * LDE + ccol] = v;
    }
    __syncthreads();

    // ---- softmax + top-8 per row: each wave owns 4 rows, 2 experts/lane ----
#pragma unroll 1
    for (int rr = 0; rr < 4; ++rr) {
        const int r = (wave << 2) + rr;
        float v0 = smem[r * LDE + lane];
        float v1 = smem[r * LDE + lane + 32];

        float m = fmaxf(v0, v1);
#pragma unroll
        for (int off = 16; off; off >>= 1) m = fmaxf(m, __shfl_xor(m, off, 32));
        const float e0 = __expf(v0 - m);
        const float e1 = __expf(v1 - m);
        float s = e0 + e1;
#pragma unroll
        for (int off = 16; off; off >>= 1) s += __shfl_xor(s, off, 32);
        const float inv = 1.0f / s;
        const float w0 = e0 * inv, w1 = e1 * inv;
        wout[(size_t)(row0 + r) * Edim + lane]      = w0;
        wout[(size_t)(row0 + r) * Edim + lane + 32] = w1;

        float l0 = w0, l1 = w1;   // weights in [0,1]; -1 = removed
#pragma unroll 1
        for (int t = 0; t < TOPKK; ++t) {
            float bv; int bi;
            if (l0 >= l1) { bv = l0; bi = lane; } else { bv = l1; bi = lane + 32; }
#pragma unroll
            for (int off = 16; off; off >>= 1) {
                const float ov = __shfl_xor(bv, off, 32);
                const int   oi = __shfl_xor(bi, off, 32);
                if (ov > bv || (ov == bv && oi < bi)) { bv = ov; bi = oi; }
            }
            if (lane == 0) iout[(size_t)(row0 + r) * TOPKK + t] = bi;
            if (bi == lane)             l0 = -1.0f;
            else if (bi == (lane + 32)) l1 = -1.0f;
        }
    }
}

extern "C" void kernel_launch(void* const* d_in, const int* in_sizes, int n_in,
                              void* d_out, int out_size, void* d_ws, size_t ws_size,
                              hipStream_t stream) {
    (void)in_sizes; (void)n_in; (void)out_size; (void)d_ws; (void)ws_size;
    const float* X = (const float*)d_in[0];   // hidden_states [T,H]
    const float* W = (const float*)d_in[1];   // router weight [E,H]
    float* logits = (float*)d_out;                            // [T,E]
    float* wout   = logits + (size_t)Tdim * Edim;             // [T,E]
    int*   iout   = (int*)(logits + (size_t)2 * Tdim * Edim); // [T,8] int32
    router_fused<<<dim3(Tdim / ROWS), 256, 0, stream>>>(X, W, logits, wout, iout);
}